// kernel_generated_0_21887153341200
// MI455X (gfx1250) — compile-verified
//
#include <hip/hip_runtime.h>

typedef __attribute__((ext_vector_type(2))) float v2f;
typedef __attribute__((ext_vector_type(8))) float v8f;

#define H 56
#define W 56
#define PIX (H * W)          // 3136 pixels = 196 * 16
#define KTOT 192             // 32 (l) * 2 (j) * 3 (k)
#define LDS_STRIDE 196       // 192 + 4 floats padding -> conflict-free fragment reads

// One block = 16-pixel M-tile, 4 waves = 4 x 16-channel N-tiles (64 channels).
__global__ void __launch_bounds__(128)
shiftconv_wmma_f32(const float* __restrict__ x,   // (64, 56, 56)
                   const float* __restrict__ w,   // (64, 32, 2, 3)
                   float* __restrict__ y)         // (64, 56, 56)
{
    __shared__ float As[16 * LDS_STRIDE];

    const int tid   = threadIdx.x;
    const int lane  = tid & 31;          // wave32
    const int wid   = tid >> 5;          // 0..3 : N-tile (channel group)
    const int mtile = blockIdx.x;        // 0..195 : pixel tile

    // ---- Stage im2col A-tile (16 pixels x 192 K) into LDS ----
    // K ordering r = l*6 + j*3 + k  (== w's innermost layout, so B loads are contiguous)
    for (int e = tid; e < 16 * KTOT; e += 128) {
        const int mm  = e / KTOT;
        const int r   = e - mm * KTOT;
        const int l   = r / 6;
        const int rem = r - l * 6;
        const int j   = rem / 3;
        const int k   = rem - j * 3;

        const int p = mtile * 16 + mm;   // global pixel index
        const int n = p / W;
        const int o = p - n * W;

        const int o2 = o + k - 1;        // unfold window (pad 1 each side)
        float val = 0.0f;
        if (o2 >= 0 && o2 < W) {
            const int o3 = (o2 + W - 1) % W;                   // roll +1 along W
            const int c  = (j == 0) ? (32 + l) : l;            // fused (56,2) roll:
            const int n2 = (j == 0) ? ((n + H - 1) % H) : n;   // j=0 -> upper half, row n-1
            val = x[c * PIX + n2 * W + o3];
        }
        As[mm * LDS_STRIDE + r] = val;
    }
    __syncthreads();

    // ---- 16x16 output tile via chained V_WMMA_F32_16X16X4_F32 over K=192 ----
    // A frag layout: lane L -> row M = L&15, K pair = 2*(L>>4) + {0,1}
    // B frag layout: lane L -> col N = L&15, same K pair
    const int half = lane >> 4;
    const int idx  = lane & 15;
    const float* bcol = w + (wid * 16 + idx) * KTOT;   // weights of output channel i

    v8f acc = {};
    #pragma unroll
    for (int kb = 0; kb < KTOT / 4; ++kb) {
        const int kbase = kb * 4 + half * 2;
        v2f a, b;
        a.x = As[idx * LDS_STRIDE + kbase];
        a.y = As[idx * LDS_STRIDE + kbase + 1];
        b.x = bcol[kbase];
        b.y = bcol[kbase + 1];
        // 8 args: (neg_a, A, neg_b, B, c_mod, C, reuse_a, reuse_b)
        acc = __builtin_amdgcn_wmma_f32_16x16x4_f32(false, a, false, b,
                                                    (short)0, acc, false, false);
    }

    // ---- Store: lane L, vgpr r holds C[M = r + 8*half, N = idx] ----
    const int i = wid * 16 + idx;        // output channel
    #pragma unroll
    for (int r = 0; r < 8; ++r) {
        const int p = mtile * 16 + r + 8 * half;
        const int n = p / W;
        const int o = p - n * W;
        y[i * PIX + n * W + o] = acc[r];
    }
}

extern "C" void kernel_launch(void* const* d_in, const int* in_sizes, int n_in,
                              void* d_out, int out_size, void* d_ws, size_t ws_size,
                              hipStream_t stream) {
    const float* x = (const float*)d_in[0];   // (1,64,56,56) fp32
    const float* w = (const float*)d_in[1];   // (64,32,2,3) fp32
    float* y = (float*)d_out;                 // (1,64,56,56) fp32

    shiftconv_wmma_f32<<<PIX / 16, 128, 0, stream>>>(x, w, y);
}